// NC_40114994544973
// MI455X (gfx1250) — compile-verified
//
#include <hip/hip_runtime.h>
#include <hip/hip_bf16.h>

#define NMAT   512
#define NBATCH 128
#define BLK    16
#define NBLK   32      // 512/16 blocks
#define NPAIR  16      // disjoint pairs per round
#define NROUND 31      // round-robin rounds per sweep
#define NSWEEP 8       // block-Jacobi sweeps

typedef float v2f __attribute__((ext_vector_type(2)));
typedef float v8f __attribute__((ext_vector_type(8)));

// Round-robin (circle method) pairing of 32 block indices.
__device__ __forceinline__ int rr_slot(int round, int i) {
  return (i == 0) ? 0 : (1 + ((i - 1 + round) % 31));
}

// ---------------------------------------------------------------------------
// S = triu(x) + triu(x)^T - diag(x)  ==  S[i][j] = x[min(i,j)][max(i,j)]
// ---------------------------------------------------------------------------
__global__ void k_symmetrize(const float* __restrict__ x, float* __restrict__ A) {
  size_t idx = (size_t)blockIdx.x * blockDim.x + threadIdx.x;
  size_t total = (size_t)NBATCH * NMAT * NMAT;
  if (idx >= total) return;
  size_t b = idx / ((size_t)NMAT * NMAT);
  int rem = (int)(idx - b * (size_t)NMAT * NMAT);
  int i = rem >> 9, j = rem & 511;
  size_t base = b * (size_t)NMAT * NMAT;
  A[idx] = (i <= j) ? x[base + (size_t)i * NMAT + j]
                    : x[base + (size_t)j * NMAT + i];
}

__global__ void k_init_eye(float* __restrict__ V) {
  size_t idx = (size_t)blockIdx.x * blockDim.x + threadIdx.x;
  size_t total = (size_t)NBATCH * NMAT * NMAT;
  if (idx >= total) return;
  size_t b = idx / ((size_t)NMAT * NMAT);
  int rem = (int)(idx - b * (size_t)NMAT * NMAT);
  int i = rem >> 9, j = rem & 511;
  V[idx] = (i == j) ? 1.0f : 0.0f;
}

// ---------------------------------------------------------------------------
// Solve 32x32 pivot subproblem: cyclic Jacobi on M = A([p;q],[p;q]),
// accumulate rotation U (32x32). One wave per (pair, batch).
// ---------------------------------------------------------------------------
__global__ __launch_bounds__(32)
void k_solve(const float* __restrict__ A, float* __restrict__ Ubuf, int round) {
  __shared__ float M[32][33];
  __shared__ float U[32][33];
  const int pair = blockIdx.x;
  const int b    = blockIdx.y;
  const int lane = threadIdx.x;
  int p = rr_slot(round, pair);
  int q = rr_slot(round, 31 - pair);
  const int pb = ((p < q) ? p : q) * BLK;
  const int qb = ((p < q) ? q : p) * BLK;
  const float* Ab = A + (size_t)b * NMAT * NMAT;

  for (int t = 0; t < 32; ++t) {
    int idx = t * 32 + lane;
    int row = idx >> 5, col = idx & 31;
    int grow = (row < BLK) ? (pb + row) : (qb + row - BLK);
    int gcol = (col < BLK) ? (pb + col) : (qb + col - BLK);
    M[row][col] = Ab[(size_t)grow * NMAT + gcol];
    U[row][col] = (row == col) ? 1.0f : 0.0f;
  }
  __syncthreads();

  for (int sweep = 0; sweep < 10; ++sweep) {
    for (int pp = 0; pp < 31; ++pp) {
      for (int qq = pp + 1; qq < 32; ++qq) {
        float apq = M[pp][qq];                 // uniform scalar reads
        if (fabsf(apq) > 1e-10f) {
          float app = M[pp][pp], aqq = M[qq][qq];
          float tau = (aqq - app) / (2.0f * apq);
          float tt  = ((tau >= 0.0f) ? 1.0f : -1.0f) /
                      (fabsf(tau) + sqrtf(1.0f + tau * tau));
          float c = rsqrtf(1.0f + tt * tt);
          float s = tt * c;
          __syncthreads();
          float rp = M[pp][lane], rq = M[qq][lane];   // rows
          M[pp][lane] = c * rp - s * rq;
          M[qq][lane] = s * rp + c * rq;
          __syncthreads();
          float cp = M[lane][pp], cq = M[lane][qq];   // cols
          M[lane][pp] = c * cp - s * cq;
          M[lane][qq] = s * cp + c * cq;
          float up = U[lane][pp], uq = U[lane][qq];   // accumulate U
          U[lane][pp] = c * up - s * uq;
          U[lane][qq] = s * up + c * uq;
          __syncthreads();
        }
      }
    }
  }

  float* Ub = Ubuf + ((size_t)b * NPAIR + pair) * 1024;
  for (int t = 0; t < 32; ++t) {
    int idx = t * 32 + lane;
    Ub[idx] = U[idx >> 5][idx & 31];
  }
}

// ---------------------------------------------------------------------------
// Row update: A([p;q], :) = U^T * A([p;q], :), in place. One wave computes a
// full 32x16 output chunk (exclusive ownership -> safe in-place).
// WMMA f32 16x16x4: A-frag lane(m = l&15, kb = (l>>4)*2), B-frag symmetric.
// ---------------------------------------------------------------------------
__global__ __launch_bounds__(32)
void k_rowupd(float* __restrict__ A, const float* __restrict__ Ubuf, int round) {
  __shared__ float U[32][33];
  __shared__ float As[32][17];
  const int nt   = blockIdx.x;    // column chunk 0..31
  const int pair = blockIdx.y;
  const int b    = blockIdx.z;
  const int lane = threadIdx.x;
  int p = rr_slot(round, pair);
  int q = rr_slot(round, 31 - pair);
  const int pb = ((p < q) ? p : q) * BLK;
  const int qb = ((p < q) ? q : p) * BLK;
  float* Ab = A + (size_t)b * NMAT * NMAT;
  const float* Ub = Ubuf + ((size_t)b * NPAIR + pair) * 1024;

  for (int t = 0; t < 32; ++t) {
    int idx = t * 32 + lane;
    U[idx >> 5][idx & 31] = Ub[idx];
  }
  for (int t = 0; t < 16; ++t) {
    int idx = t * 32 + lane;          // 512 elements: 32 rows x 16 cols
    int row = idx >> 4, col = idx & 15;
    int grow = (row < BLK) ? (pb + row) : (qb + row - BLK);
    As[row][col] = Ab[(size_t)grow * NMAT + nt * BLK + col];
  }
  __syncthreads();

  const int m  = lane & 15;
  const int kb = (lane >> 4) * 2;
  v8f acc[2] = {};
#pragma unroll
  for (int mt = 0; mt < 2; ++mt) {
#pragma unroll
    for (int kk = 0; kk < 32; kk += 4) {
      v2f a, bv;
      a.x  = U[kk + kb][mt * 16 + m];      // (U^T)(m,k) = U[k][m]
      a.y  = U[kk + kb + 1][mt * 16 + m];
      bv.x = As[kk + kb][m];
      bv.y = As[kk + kb + 1][m];
      acc[mt] = __builtin_amdgcn_wmma_f32_16x16x4_f32(
          false, a, false, bv, (short)0, acc[mt], false, false);
    }
  }

  const int n  = lane & 15;
  const int mh = (lane >> 4) * 8;
#pragma unroll
  for (int mt = 0; mt < 2; ++mt) {
#pragma unroll
    for (int r = 0; r < 8; ++r) {
      int row  = mt * 16 + mh + r;
      int grow = (row < BLK) ? (pb + row) : (qb + row - BLK);
      Ab[(size_t)grow * NMAT + nt * BLK + n] = acc[mt][r];
    }
  }
}

// ---------------------------------------------------------------------------
// Column update, in place, applied to BOTH A and eigenvector accumulator V:
// X(:, [p;q]) = X(:, [p;q]) * U.  One wave owns a 16-row strip (exclusive).
// ---------------------------------------------------------------------------
__global__ __launch_bounds__(32)
void k_colupd(float* __restrict__ A, float* __restrict__ V,
              const float* __restrict__ Ubuf, int round) {
  __shared__ float U[32][33];
  __shared__ float Xs[16][33];
  const int mt   = blockIdx.x;    // row strip 0..31
  const int pair = blockIdx.y;
  const int b    = blockIdx.z;
  const int lane = threadIdx.x;
  int p = rr_slot(round, pair);
  int q = rr_slot(round, 31 - pair);
  const int pb = ((p < q) ? p : q) * BLK;
  const int qb = ((p < q) ? q : p) * BLK;
  const float* Ub = Ubuf + ((size_t)b * NPAIR + pair) * 1024;

  for (int t = 0; t < 32; ++t) {
    int idx = t * 32 + lane;
    U[idx >> 5][idx & 31] = Ub[idx];
  }

  float* mats[2] = {A, V};
  for (int which = 0; which < 2; ++which) {
    float* Xb = mats[which] + (size_t)b * NMAT * NMAT;
    __syncthreads();
    for (int t = 0; t < 16; ++t) {
      int idx = t * 32 + lane;        // 512 elements: 16 rows x 32 cols
      int row = idx >> 5, col = idx & 31;
      int gcol = (col < BLK) ? (pb + col) : (qb + col - BLK);
      Xs[row][col] = Xb[(size_t)(mt * BLK + row) * NMAT + gcol];
    }
    __syncthreads();

    const int m  = lane & 15;
    const int kb = (lane >> 4) * 2;
    v8f acc[2] = {};
#pragma unroll
    for (int nt2 = 0; nt2 < 2; ++nt2) {
#pragma unroll
      for (int kk = 0; kk < 32; kk += 4) {
        v2f a, bv;
        a.x  = Xs[m][kk + kb];
        a.y  = Xs[m][kk + kb + 1];
        bv.x = U[kk + kb][nt2 * 16 + m];
        bv.y = U[kk + kb + 1][nt2 * 16 + m];
        acc[nt2] = __builtin_amdgcn_wmma_f32_16x16x4_f32(
            false, a, false, bv, (short)0, acc[nt2], false, false);
      }
    }
    __syncthreads();

    const int n  = lane & 15;
    const int mh = (lane >> 4) * 8;
#pragma unroll
    for (int nt2 = 0; nt2 < 2; ++nt2) {
      int col  = nt2 * 16 + n;
      int gcol = (col < BLK) ? (pb + col) : (qb + col - BLK);
#pragma unroll
      for (int r = 0; r < 8; ++r) {
        int row = mt * BLK + mh + r;
        Xb[(size_t)row * NMAT + gcol] = acc[nt2][r];
      }
    }
  }
}

// ---------------------------------------------------------------------------
// Output: eigenvalues = diag(A); ascending rank-sort; out[b][k] = V[1][idx_k],
// sign-fixed by element at rank 0 (jnp.sign semantics: sign(0) == 0).
// ---------------------------------------------------------------------------
__global__ __launch_bounds__(NMAT)
void k_output(const float* __restrict__ A, const float* __restrict__ V,
              float* __restrict__ out) {
  __shared__ float d[NMAT];
  __shared__ float yb[NMAT];
  const int b = blockIdx.x;
  const int j = threadIdx.x;
  d[j] = A[((size_t)b * NMAT + j) * NMAT + j];
  __syncthreads();
  float dj = d[j];
  int rank = 0;
  for (int mI = 0; mI < NMAT; ++mI) {
    float dm = d[mI];
    rank += (dm < dj || (dm == dj && mI < j)) ? 1 : 0;
  }
  yb[rank] = V[((size_t)b * NMAT + 1) * NMAT + j];   // row 1 of eigvec matrix
  __syncthreads();
  float y0 = yb[0];
  float sgn = (y0 > 0.0f) ? 1.0f : ((y0 < 0.0f) ? -1.0f : 0.0f);
  out[(size_t)b * NMAT + j] = yb[j] * sgn;
}

// ---------------------------------------------------------------------------
extern "C" void kernel_launch(void* const* d_in, const int* in_sizes, int n_in,
                              void* d_out, int out_size, void* d_ws, size_t ws_size,
                              hipStream_t stream) {
  (void)in_sizes; (void)n_in; (void)out_size; (void)ws_size;
  const float* x = (const float*)d_in[0];
  float* out = (float*)d_out;

  char* ws = (char*)d_ws;
  const size_t matbytes = (size_t)NBATCH * NMAT * NMAT * sizeof(float);
  float* A    = (float*)ws; ws += matbytes;                                   // 134 MB
  float* V    = (float*)ws; ws += matbytes;                                   // 134 MB
  float* Ubuf = (float*)ws; ws += (size_t)NBATCH * NPAIR * 1024 * sizeof(float); // 8 MB

  {
    size_t total = (size_t)NBATCH * NMAT * NMAT;
    int bs = 256;
    int nb = (int)((total + bs - 1) / bs);
    k_symmetrize<<<nb, bs, 0, stream>>>(x, A);
    k_init_eye<<<nb, bs, 0, stream>>>(V);
  }

  dim3 gS(NPAIR, NBATCH);
  dim3 gU(NBLK, NPAIR, NBATCH);
  dim3 bW(32);
  for (int sweep = 0; sweep < NSWEEP; ++sweep) {
    for (int round = 0; round < NROUND; ++round) {
      k_solve <<<gS, bW, 0, stream>>>(A, Ubuf, round);
      k_rowupd<<<gU, bW, 0, stream>>>(A, Ubuf, round);
      k_colupd<<<gU, bW, 0, stream>>>(A, V, Ubuf, round);
    }
  }

  k_output<<<NBATCH, NMAT, 0, stream>>>(A, V, out);
}